// VanillaRNN_4922032521711
// MI455X (gfx1250) — compile-verified
//
#include <hip/hip_runtime.h>
#include <hip/hip_bf16.h>
#include <math.h>

#define VOCAB 32000
#define HID   256
#define BATCH 16
#define SEQ   256

typedef __attribute__((ext_vector_type(2))) float v2f;
typedef __attribute__((ext_vector_type(8))) float v8f;

__device__ __forceinline__ v8f wmma_f32_k4(v2f a, v2f b, v8f c) {
    // D = A(16x4 f32) x B(4x16 f32) + C(16x16 f32)
    return __builtin_amdgcn_wmma_f32_16x16x4_f32(
        /*neg_a=*/false, a, /*neg_b=*/false, b,
        /*c_mod=*/(short)0, c, /*reuse_a=*/false, /*reuse_b=*/false);
}

// -----------------------------------------------------------------------------
// Kernel 1: xproj[s*16+b][n] = sum_k emb[x[b][s]][k] * Wxh[n][k] + bxh[n]
// One block (512 thr = 16 waves) per timestep s; wave w owns N-tile w.
// -----------------------------------------------------------------------------
__global__ __launch_bounds__(512) void k_embed_xproj(
    const int*   __restrict__ x,     // [B,S]
    const float* __restrict__ emb,   // [V,H]
    const float* __restrict__ Wxh,   // [H,H]
    const float* __restrict__ bxh,   // [H]
    float*       __restrict__ xproj) // [S,B,H] flattened (s*16+b)*H + n
{
    __shared__ float A[BATCH][HID];          // 16 KB gathered embedding panel
    const int s   = blockIdx.x;
    const int tid = threadIdx.x;

    for (int i = tid; i < BATCH * HID; i += 512) {
        const int b = i / HID, k = i % HID;
        const int tok = x[b * SEQ + s];
        A[b][k] = emb[(size_t)tok * HID + k];
    }
    __syncthreads();

    const int wave  = tid >> 5;              // N-tile index 0..15
    const int lane  = tid & 31;
    const int row   = lane & 15;             // M (A) / N (B,C) row within tile
    const int khalf = (lane >> 4) * 2;       // K sub-offset 0 or 2
    const int n     = wave * 16 + row;

    v8f c = {};
    #pragma unroll 8
    for (int kk = 0; kk < HID; kk += 4) {
        v2f a, b2;
        a.x  = A[row][kk + khalf];
        a.y  = A[row][kk + khalf + 1];
        b2.x = Wxh[n * HID + kk + khalf];     // B[k][n] = Wxh[n][k]
        b2.y = Wxh[n * HID + kk + khalf + 1];
        c = wmma_f32_k4(a, b2, c);
    }

    const float bias = bxh[n];
    #pragma unroll
    for (int r = 0; r < 8; ++r) {
        const int brow = r + (lane >> 4) * 8;  // batch row in tile
        xproj[(s * BATCH + brow) * HID + n] = c[r] + bias;
    }
}

// -----------------------------------------------------------------------------
// Kernel 2: sequential scan, single workgroup (16 waves on one WGP).
// h_t = tanh(xproj_t + h_{t-1} @ Whh^T + bhh); Whh fragments live in VGPRs.
// -----------------------------------------------------------------------------
__global__ __launch_bounds__(512) void k_rnn_scan(
    const float* __restrict__ xproj, // [S,B,H]
    const float* __restrict__ Whh,   // [H,H]
    const float* __restrict__ bhh,   // [H]
    float*       __restrict__ hs)    // [S,B,H]
{
    __shared__ float hbuf[2][BATCH][HID];    // 32 KB ping-pong hidden state
    const int tid   = threadIdx.x;
    const int wave  = tid >> 5;
    const int lane  = tid & 31;
    const int row   = lane & 15;
    const int khalf = (lane >> 4) * 2;
    const int n     = wave * 16 + row;

    // Register-resident Whh B-fragments for this wave's N-tile: 64 x v2f.
    v2f bfrag[64];
    #pragma unroll
    for (int ks = 0; ks < 64; ++ks) {
        const int k = ks * 4 + khalf;
        bfrag[ks].x = Whh[n * HID + k];
        bfrag[ks].y = Whh[n * HID + k + 1];
    }
    const float bias = bhh[n];

    // h0 = 0
    for (int i = tid; i < BATCH * HID; i += 512) ((float*)hbuf[0])[i] = 0.0f;
    __syncthreads();

    for (int t = 0; t < SEQ; ++t) {
        const int cur = t & 1, nxt = cur ^ 1;
        v8f c = {};
        #pragma unroll
        for (int ks = 0; ks < 64; ++ks) {
            const int k = ks * 4 + khalf;
            v2f a;
            a.x = hbuf[cur][row][k];
            a.y = hbuf[cur][row][k + 1];
            c = wmma_f32_k4(a, bfrag[ks], c);
        }
        #pragma unroll
        for (int r = 0; r < 8; ++r) {
            const int brow = r + (lane >> 4) * 8;
            const float v = tanhf(c[r] + xproj[(t * BATCH + brow) * HID + n] + bias);
            hbuf[nxt][brow][n] = v;
            hs[(t * BATCH + brow) * HID + n] = v;
        }
        __syncthreads();
    }
}

// -----------------------------------------------------------------------------
// Kernel 3: out[b][s][v] = sum_k hs[s*16+b][k] * Wfc[v][k] + bfc[v]
// Block: 16 waves, one timestep (M-tile) x 16 V-tiles; A panel in LDS.
// Memory bound: 512 MB store @ 23.3 TB/s ~ 22 us; Wfc (33 MB) is L2-resident.
// -----------------------------------------------------------------------------
__global__ __launch_bounds__(512) void k_fc_out(
    const float* __restrict__ hs,   // [S,B,H]
    const float* __restrict__ Wfc,  // [V,H]
    const float* __restrict__ bfc,  // [V]
    float*       __restrict__ out)  // [B,S,V]
{
    __shared__ float A[BATCH][HID];          // 16 KB hidden-state panel
    const int s   = blockIdx.y;              // timestep == M-tile
    const int tid = threadIdx.x;

    for (int i = tid; i < BATCH * HID; i += 512)
        ((float*)A)[i] = hs[s * BATCH * HID + i];
    __syncthreads();

    const int wave  = tid >> 5;
    const int lane  = tid & 31;
    const int row   = lane & 15;
    const int khalf = (lane >> 4) * 2;
    const int vt    = blockIdx.x * 16 + wave;   // V-tile 0..1999
    const int v     = vt * 16 + row;

    v8f c = {};
    #pragma unroll 8
    for (int kk = 0; kk < HID; kk += 4) {
        v2f a, b2;
        a.x  = A[row][kk + khalf];
        a.y  = A[row][kk + khalf + 1];
        b2.x = Wfc[(size_t)v * HID + kk + khalf];  // B[k][v] = Wfc[v][k]
        b2.y = Wfc[(size_t)v * HID + kk + khalf + 1];
        c = wmma_f32_k4(a, b2, c);
    }

    const float bias = bfc[v];
    #pragma unroll
    for (int r = 0; r < 8; ++r) {
        const int b = r + (lane >> 4) * 8;       // batch row
        out[((size_t)b * SEQ + s) * VOCAB + v] = c[r] + bias;
    }
}

// -----------------------------------------------------------------------------
extern "C" void kernel_launch(void* const* d_in, const int* in_sizes, int n_in,
                              void* d_out, int out_size, void* d_ws, size_t ws_size,
                              hipStream_t stream) {
    const int*   x    = (const int*)  d_in[0];
    const float* emb  = (const float*)d_in[1];
    const float* Wxh  = (const float*)d_in[2];
    const float* bxh  = (const float*)d_in[3];
    const float* Whh  = (const float*)d_in[4];
    const float* bhh  = (const float*)d_in[5];
    const float* Wfc  = (const float*)d_in[6];
    const float* bfc  = (const float*)d_in[7];
    float* out = (float*)d_out;

    float* xproj = (float*)d_ws;                     // [S*B*H] = 4 MB
    float* hs    = xproj + (size_t)SEQ * BATCH * HID; // [S*B*H] = 4 MB

    k_embed_xproj<<<SEQ, 512, 0, stream>>>(x, emb, Wxh, bxh, xproj);
    k_rnn_scan  <<<1,   512, 0, stream>>>(xproj, Whh, bhh, hs);
    k_fc_out    <<<dim3(VOCAB / 256, SEQ), 512, 0, stream>>>(hs, Wfc, bfc, out);
}